// Detector_43834436223701
// MI455X (gfx1250) — compile-verified
//
#include <hip/hip_runtime.h>
#include <stdint.h>

typedef __attribute__((ext_vector_type(4))) unsigned int u32x4;
typedef __attribute__((ext_vector_type(8))) int          i32x8;
typedef __attribute__((ext_vector_type(4))) int          i32x4;
typedef __attribute__((ext_vector_type(2))) float        f32x2;

#ifndef __has_builtin
#define __has_builtin(x) 0
#endif

// YOLO decode: one thread per (b, anchor, spatial cell).
// tid = (b*3 + a)*HW + s  -> for fixed channel k, consecutive lanes read
// consecutive addresses (base + k*HW + s): fully coalesced 128B/wave streams.
__global__ __launch_bounds__(256) void yolo_decode_kernel(
    const float* __restrict__ in,       // [B, 255, H, W]
    const float* __restrict__ anchors,  // [3, 2]
    float* __restrict__ out,            // [B*HW*3, 6] slice of global output
    int B, int HW, int W, float tscale /* t/416 */, float inv_case /* 1/416 */)
{
  __shared__ float sAnch[8];

#if __has_builtin(__builtin_amdgcn_tensor_load_to_lds) && __has_builtin(__builtin_amdgcn_s_wait_tensorcnt)
  // Broadcast the 6 anchor floats into LDS via the Tensor Data Mover.
  // Wave 0 issues one TDM descriptor (EXEC ignored; operands wave-uniform).
  if (threadIdx.x < 32) {
    uint32_t lds_off = (uint32_t)(uintptr_t)(&sAnch[0]);   // low 32 bits of generic LDS ptr == LDS byte address
    uint64_t ga      = (uint64_t)(uintptr_t)anchors;

    u32x4 g0;
    g0[0] = 1u;                                   // count=1 (valid user descriptor), is_restore=0, gather off
    g0[1] = lds_off;                              // lds_addr (bytes)
    g0[2] = (uint32_t)ga;                         // global_addr[31:0]
    g0[3] = (uint32_t)(ga >> 32) | (2u << 30);    // global_addr[56:32] | type=2 ("image")

    i32x8 g1;
    g1[0] = 0x00020000;   // workgroup_mask=0 (not in cluster), data_size=2 -> 4 bytes
    g1[1] = (6 << 16);    // tensor_dim0 = 6 (elements)
    g1[2] = (1 << 16);    // tensor_dim1 = 1
    g1[3] = (6 << 16);    // tile_dim0 = 6
    g1[4] = 1;            // tile_dim1 = 1, tile_dim2 = 0 (unused)
    g1[5] = 6;            // tensor_dim0_stride = 6
    g1[6] = 0;
    g1[7] = 0;            // tensor_dim1_stride = 0 (single line)

    i32x4 gz4 = {0, 0, 0, 0};                     // groups 2/3: dims beyond 2D unused
    i32x8 gz8 = {0, 0, 0, 0, 0, 0, 0, 0};         // trailing group (disabled)
    __builtin_amdgcn_tensor_load_to_lds(g0, g1, gz4, gz4, gz8, 0);
    __builtin_amdgcn_s_wait_tensorcnt(0);
  }
#else
  if (threadIdx.x < 6) sAnch[threadIdx.x] = anchors[threadIdx.x];
#endif
  __syncthreads();

  const int tid   = blockIdx.x * blockDim.x + threadIdx.x;
  const int total = B * 3 * HW;
  if (tid >= total) return;

  const int s  = tid % HW;
  const int ba = tid / HW;
  const int a  = ba % 3;
  const int b  = ba / 3;

  const float* p = in + ((size_t)(b * 255 + a * 85)) * (size_t)HW + (size_t)s;

  // Channels 0..4: objectness + box regressors (single-use data -> NT hints)
  const float t0 = __builtin_nontemporal_load(p);
  const float o1 = __builtin_nontemporal_load(p + (size_t)HW);
  const float o2 = __builtin_nontemporal_load(p + 2 * (size_t)HW);
  const float o3 = __builtin_nontemporal_load(p + 3 * (size_t)HW);
  const float o4 = __builtin_nontemporal_load(p + 4 * (size_t)HW);

  // First-occurrence argmax over 80 class logits. 4 blocked accumulators
  // (contiguous k-ranges) combined in order with strict '>' preserves the
  // jnp.argmax first-index tie rule while giving 4-way ILP.
  const float* pc = p + 5 * (size_t)HW;
  float bv[4];
  int   bi[4];
#pragma unroll
  for (int blk = 0; blk < 4; ++blk) {
    float bestv = __builtin_nontemporal_load(pc + (size_t)(blk * 20) * (size_t)HW);
    int   besti = blk * 20;
#pragma unroll
    for (int k = 1; k < 20; ++k) {
      const int   c = blk * 20 + k;
      const float v = __builtin_nontemporal_load(pc + (size_t)c * (size_t)HW);
      if (v > bestv) { bestv = v; besti = c; }
    }
    bv[blk] = bestv;
    bi[blk] = besti;
  }
  float bestv = bv[0];
  int   besti = bi[0];
#pragma unroll
  for (int blk = 1; blk < 4; ++blk)
    if (bv[blk] > bestv) { bestv = bv[blk]; besti = bi[blk]; }

  const float conf = 1.0f / (1.0f + expf(-t0));
  const bool  keep = conf > 0.5f;   // exact reference semantics (incl. rounding edge)

  const int hy = s / W;
  const int wx = s - hy * W;

  const float cx = ((float)wx + o1) * tscale;
  const float cy = ((float)hy + o2) * tscale;
  const float aw = sAnch[a * 2 + 0];
  const float ah = sAnch[a * 2 + 1];
  const float bw = aw * expf(o3) * inv_case;
  const float bh = ah * expf(o4) * inv_case;
  const float cl = (float)besti;

  // Output row ordering matches reshape(B,H,W,3,6) -> row = (b*HW + s)*3 + a
  const int row = (b * HW + s) * 3 + a;
  float* q = out + (size_t)row * 6;

  f32x2 v01, v23, v45;
  v01.x = keep ? conf : 0.0f;
  v01.y = keep ? cx   : 0.0f;
  v23.x = keep ? cy   : 0.0f;
  v23.y = keep ? bw   : 0.0f;
  v45.x = keep ? bh   : 0.0f;
  v45.y = keep ? cl   : 0.0f;

  f32x2* q2 = (f32x2*)q;            // row*24B is 8B-aligned
  __builtin_nontemporal_store(v01, q2 + 0);
  __builtin_nontemporal_store(v23, q2 + 1);
  __builtin_nontemporal_store(v45, q2 + 2);
}

extern "C" void kernel_launch(void* const* d_in, const int* in_sizes, int n_in,
                              void* d_out, int out_size, void* d_ws, size_t ws_size,
                              hipStream_t stream) {
  (void)n_in; (void)out_size; (void)d_ws; (void)ws_size;

  const float* feats[3]   = { (const float*)d_in[0], (const float*)d_in[1], (const float*)d_in[2] };
  const float* anchors[3] = { (const float*)d_in[3], (const float*)d_in[4], (const float*)d_in[5] };
  const int    Wd[3]      = { 13, 26, 52 };
  const float  tv[3]      = { 32.0f, 16.0f, 8.0f };

  float* out = (float*)d_out;
  size_t off = 0;

  for (int i = 0; i < 3; ++i) {
    const int HW    = Wd[i] * Wd[i];
    const int B     = in_sizes[i] / (255 * HW);
    const int total = B * 3 * HW;
    const int blocks = (total + 255) / 256;
    yolo_decode_kernel<<<blocks, 256, 0, stream>>>(
        feats[i], anchors[i], out + off, B, HW, Wd[i],
        tv[i] / 416.0f, 1.0f / 416.0f);
    off += (size_t)total * 6;
  }
}